// LSTM_31413390803182
// MI455X (gfx1250) — compile-verified
//
#include <hip/hip_runtime.h>
#include <math.h>

typedef __attribute__((ext_vector_type(2))) float v2f;
typedef __attribute__((ext_vector_type(8))) float v8f;

#define KB 64                  // batch
#define KT 512                 // timesteps
#define KI 256                 // input size
#define KH 512                 // hidden size
#define KO 256                 // output size
#define JPW 8                  // hidden units per recurrence WG
#define NSL (4 * JPW)          // gate rows per WG (i,f,g,o x 8) = 32
#define NWG (KH / JPW)         // recurrence workgroups = 64
#define WLDA (KH + KI + 4)     // LDS weight-slice row stride (772) for bank spread
#define GLD (NSL + 1)          // LDS gate-tile stride (33) -> conflict-free reads

// Slabs hT/xT use pair-interleaved K-major layout: element (k, b) lives at
//   slab[((k >> 1) * KB + b) * 2 + (k & 1)]
// so a lane's WMMA A-fragment (k, k+1) is one aligned 8-byte global_load_b64,
// and lanes 0..15 cover a fully contiguous 128B segment.
#define PAIR_OFF(k, b) ((((k) >> 1) * KB + (b)) * 2 + ((k) & 1))

// ---------------------------------------------------------------------------
__global__ void lstm_init_sync(int* c) { *c = 0; }

// ---------------------------------------------------------------------------
// One-time transpose: x[B][T][I] -> xT[T] pair-interleaved [I/2][B][2].
// ---------------------------------------------------------------------------
__global__ void __launch_bounds__(256)
xpose_kernel(const float* __restrict__ x, float* __restrict__ xT) {
  __shared__ float tile[64][65];
  const int t   = blockIdx.x;
  const int i0  = blockIdx.y * 64;
  const int tid = threadIdx.x;
#pragma unroll
  for (int e = 0; e < 16; ++e) {
    const int idx = tid + e * 256;
    const int b = idx >> 6, ii = idx & 63;
    tile[b][ii] = x[((size_t)b * KT + t) * KI + i0 + ii];
  }
  __syncthreads();
  float* dst = xT + (size_t)t * KI * KB;
#pragma unroll
  for (int e = 0; e < 16; ++e) {
    const int idx = tid + e * 256;
    const int ii = idx >> 6, b = idx & 63;
    dst[PAIR_OFF(i0 + ii, b)] = tile[b][ii];
  }
}

// ---------------------------------------------------------------------------
// Persistent recurrence. 64 WGs x 256 threads (8 waves). WG owns hidden slice
// j0 = wg*8 (32 gate rows) for all 64 batch rows. Per step:
//   gates[64x32] = [h_{t-1} | x_t][64x768] @ Wslice^T      (f32 WMMA)
// A-fragments are direct b64 global loads (pair-interleaved slabs); the
// weight slice (96KB) is LDS-resident for the whole T loop. Cross-WG h
// coupling via monotone spin barrier over the history buffer (no WAR hazard
// since full history is kept — it feeds the FC kernel anyway).
// ---------------------------------------------------------------------------
__global__ void __launch_bounds__(256)
lstm_recur_kernel(const float* __restrict__ xT,    // [T] pair slabs of x
                  const float* __restrict__ W_ih,  // [4H][I]
                  const float* __restrict__ W_hh,  // [4H][H]
                  const float* __restrict__ b_ih,  // [4H]
                  const float* __restrict__ b_hh,  // [4H]
                  float* __restrict__ hT,          // [T] pair slabs of h
                  int* sync_counter)
{
  extern __shared__ char smem_raw[];
  float* Wl = (float*)smem_raw;          // [NSL][WLDA] resident weight slice
  float* Gl = Wl + NSL * WLDA;           // [KB][GLD]   gate pre-activations
  float* bl = Gl + KB * GLD;             // [NSL]       fused biases

  const int tid = threadIdx.x;
  const int wg  = blockIdx.x;
  const int j0  = wg * JPW;

  // one-time weight gather: rows packed [i8|f8|g8|o8], K = [h(512) | x(256)]
  for (int idx = tid; idx < NSL * (KH + KI); idx += 256) {
    const int r = idx / (KH + KI), k = idx - r * (KH + KI);
    const int g = r >> 3, j = j0 + (r & 7);
    const int row = g * KH + j;
    Wl[r * WLDA + k] = (k < KH) ? W_hh[(size_t)row * KH + k]
                                : W_ih[(size_t)row * KI + (k - KH)];
  }
  if (tid < NSL) {
    const int g = tid >> 3, j = j0 + (tid & 7);
    bl[tid] = b_ih[g * KH + j] + b_hh[g * KH + j];
  }

  // WMMA tiling: 8 waves = 4 M-tiles x 2 N-tiles of 16x16
  const int wave = tid >> 5, lane = tid & 31;
  const int mt = wave >> 1, nt = wave & 1;
  const int lh = lane >> 4, l15 = lane & 15;
  const int m    = mt * 16 + l15;        // batch row this lane touches
  const int nCol = nt * 16 + l15;        // gate column this lane touches
  const int aoff = (lh * KB + m) * 2;    // lane offset within a pair slab
  const float* Bp = Wl + nCol * WLDA + lh * 2;   // LDS B-fragment base

  float cst[2] = {0.f, 0.f};             // cell state: pairs p=tid, tid+256

  __syncthreads();

  for (int t = 0; t < KT; ++t) {
    // ---- grid-wide release/acquire: all WGs finished step t-1 --------------
    if (t > 0) {
      __threadfence();
      if (tid == 0) {
        atomicAdd(sync_counter, 1);
        const int target = NWG * t;
        while (__hip_atomic_load(sync_counter, __ATOMIC_RELAXED,
                                 __HIP_MEMORY_SCOPE_AGENT) < target) { }
      }
      __syncthreads();
      __threadfence();
    }

    // prefetch next step's xT slab (2 cachelines/thread covers 64KB exactly)
    if (t + 1 < KT) {
      const float* xn = xT + (size_t)(t + 1) * KI * KB + tid * 64;
      __builtin_prefetch(xn, 0, 3);
      __builtin_prefetch(xn + 32, 0, 3);
    }

    v8f acc = {};
    // h-part of K (512): A = hT[t-1]; skipped at t==0 since h0 == 0
    if (t > 0) {
      const float* Ab = hT + (size_t)(t - 1) * KH * KB + aoff;
#pragma unroll 8
      for (int k0 = 0; k0 < KH; k0 += 4) {
        const v2f a = *(const v2f*)(Ab + (size_t)k0 * KB);
        const v2f b = *(const v2f*)(Bp + k0);
        acc = __builtin_amdgcn_wmma_f32_16x16x4_f32(false, a, false, b,
                                                    (short)0, acc, false, false);
      }
    }
    // x-part of K (256): A = xT[t]; weight columns offset by KH
    {
      const float* Ab = xT + (size_t)t * KI * KB + aoff;
      const float* Bx = Bp + KH;
#pragma unroll 8
      for (int k0 = 0; k0 < KI; k0 += 4) {
        const v2f a = *(const v2f*)(Ab + (size_t)k0 * KB);
        const v2f b = *(const v2f*)(Bx + k0);
        acc = __builtin_amdgcn_wmma_f32_16x16x4_f32(false, a, false, b,
                                                    (short)0, acc, false, false);
      }
    }

    // spill gates (+bias) to LDS: Gl[m][n] stride 33
#pragma unroll
    for (int r = 0; r < 8; ++r) {
      const int mm = mt * 16 + lh * 8 + r;   // C/D layout: lanes 16-31 = M+8
      Gl[mm * GLD + nCol] = acc[r] + bl[nCol];
    }
    __syncthreads();

    // LSTM cell update; thread owns (b,jj) pairs p=tid, tid+256
    float* hout = hT + (size_t)t * KH * KB;
#pragma unroll
    for (int q = 0; q < 2; ++q) {
      const int p  = tid + q * 256;
      const int jj = p >> 6, b = p & 63;
      const float gi = Gl[b * GLD + 0 * JPW + jj];
      const float gf = Gl[b * GLD + 1 * JPW + jj];
      const float gc = Gl[b * GLD + 2 * JPW + jj];
      const float go = Gl[b * GLD + 3 * JPW + jj];
      const float si = 1.f / (1.f + __expf(-gi));
      const float sf = 1.f / (1.f + __expf(-gf));
      const float tg = tanhf(gc);
      const float so = 1.f / (1.f + __expf(-go));
      const float cn = sf * cst[q] + si * tg;
      cst[q] = cn;
      hout[PAIR_OFF(j0 + jj, b)] = so * tanhf(cn);
    }
    __syncthreads();
  }
}

// ---------------------------------------------------------------------------
// Final FC + ReLU over the history: out[b][t][:] = relu(h_t[b] @ W_fc^T + b_fc)
// hT slabs are pair-interleaved: A-fragments are single b64 loads; each
// block.x is one t (rows = b 0..63), fully coalesced.
// ---------------------------------------------------------------------------
__global__ void __launch_bounds__(256)
fc_relu_kernel(const float* __restrict__ hT,    // [T] pair slabs of h
               const float* __restrict__ W_fc,  // [O][H]
               const float* __restrict__ b_fc,  // [O]
               float* __restrict__ out)         // [B][T][O]
{
  const int tid = threadIdx.x;
  const int wave = tid >> 5, lane = tid & 31;
  const int mt = wave >> 1, nh = wave & 1;
  const int lh = lane >> 4, l15 = lane & 15;
  const int t = blockIdx.x;
  const int colBase = blockIdx.y * 64 + nh * 32;

  const float* Ab = hT + (size_t)t * KH * KB + (lh * KB + mt * 16 + l15) * 2;
  const float* B0 = W_fc + (size_t)(colBase + l15) * KH + lh * 2;
  const float* B1 = B0 + (size_t)16 * KH;

  v8f acc0 = {}, acc1 = {};
#pragma unroll 8
  for (int k0 = 0; k0 < KH; k0 += 4) {
    const v2f a  = *(const v2f*)(Ab + (size_t)k0 * KB);
    const v2f b0 = *(const v2f*)(B0 + k0);
    const v2f b1 = *(const v2f*)(B1 + k0);
    acc0 = __builtin_amdgcn_wmma_f32_16x16x4_f32(false, a, false, b0,
                                                 (short)0, acc0, false, false);
    acc1 = __builtin_amdgcn_wmma_f32_16x16x4_f32(false, a, false, b1,
                                                 (short)0, acc1, false, false);
  }

  const int n0 = colBase + l15, n1 = n0 + 16;
  const float bias0 = b_fc[n0], bias1 = b_fc[n1];
#pragma unroll
  for (int r = 0; r < 8; ++r) {
    const int b = mt * 16 + lh * 8 + r;
    out[((size_t)b * KT + t) * KO + n0] = fmaxf(acc0[r] + bias0, 0.f);
    out[((size_t)b * KT + t) * KO + n1] = fmaxf(acc1[r] + bias1, 0.f);
  }
}

// ---------------------------------------------------------------------------
extern "C" void kernel_launch(void* const* d_in, const int* in_sizes, int n_in,
                              void* d_out, int out_size, void* d_ws, size_t ws_size,
                              hipStream_t stream) {
  (void)in_sizes; (void)n_in; (void)out_size; (void)ws_size;
  const float* x    = (const float*)d_in[0];
  const float* W_ih = (const float*)d_in[1];
  const float* W_hh = (const float*)d_in[2];
  const float* b_ih = (const float*)d_in[3];
  const float* b_hh = (const float*)d_in[4];
  const float* W_fc = (const float*)d_in[5];
  const float* b_fc = (const float*)d_in[6];

  // workspace: [0..4)           sync counter
  //            [256 .. +64MB)   hT  pair slabs, T*H*B floats
  //            [+64MB .. +96MB) xT  pair slabs, T*I*B floats
  int*   sync = (int*)d_ws;
  float* hT   = (float*)((char*)d_ws + 256);
  float* xT   = hT + (size_t)KT * KH * KB;

  lstm_init_sync<<<1, 1, 0, stream>>>(sync);
  xpose_kernel<<<dim3(KT, KI / 64), dim3(256), 0, stream>>>(x, xT);

  const size_t lds_bytes =
      (size_t)(NSL * WLDA + KB * GLD + NSL) * sizeof(float);  // ~105 KB
  lstm_recur_kernel<<<dim3(NWG), dim3(256), lds_bytes, stream>>>(
      xT, W_ih, W_hh, b_ih, b_hh, hT, sync);

  fc_relu_kernel<<<dim3(KT, KO / 64), dim3(256), 0, stream>>>(
      hT, W_fc, b_fc, (float*)d_out);
}